// GNN_386547057000
// MI455X (gfx1250) — compile-verified
//
#include <hip/hip_runtime.h>
#include <hip/hip_bf16.h>

// ---------------------------------------------------------------------------
// Types for WMMA f32 16x16x4: A is 16x4 (2 f32/lane), B is 4x16 (2 f32/lane),
// C/D is 16x16 (8 f32/lane).
// ---------------------------------------------------------------------------
typedef float v2f __attribute__((ext_vector_type(2)));
typedef float v8f __attribute__((ext_vector_type(8)));

#define HDIM 128          // hidden = feature dim
#define BM   128          // rows per block
#define KC   32           // K chunk staged in LDS (8 WMMA k-steps)
#define AS_STRIDE 36      // 32 + 4 pad -> 16-row column reads hit distinct banks

// ---------------------------------------------------------------------------
// Fused GEMM: D[M,128] = (RELU_IN? relu(A) : A)[M,128] @ B[128,128]
//                        (+ bias) (+ Cin) (then optional relu)
// 256 threads = 8 waves; waves tiled 4(M) x 2(N); each wave owns 32 rows x
// 64 cols = 8 WMMA accumulators. B is staged quad-packed (BsQ[ks][col][4])
// so every A/B fragment is one aligned b64 LDS read -> no packing movs.
// Per k-step per wave: 6 LDS b64 loads feed 8 v_wmma_f32_16x16x4_f32.
// ---------------------------------------------------------------------------
template<bool RELU_IN, bool BIAS, bool ACCUM, bool RELU_OUT>
__global__ __launch_bounds__(256)
void gemm128_wmma(const float* __restrict__ A, const float* __restrict__ B,
                  const float* __restrict__ bias, const float* __restrict__ Cin,
                  float* __restrict__ D, int M)
{
    __shared__ float As[BM][AS_STRIDE];          // 128 x 36 = 18 KB
    __shared__ float BsQ[KC / 4][HDIM][4];       // 8 x 128 x 4 = 16 KB

    const int tid       = threadIdx.x;
    const int block_row = blockIdx.x * BM;
    const int wave      = tid >> 5;
    const int lane      = tid & 31;
    const int wm        = wave & 3;          // M tile: rows wm*32
    const int wn        = wave >> 2;         // N half: cols wn*64
    const int lr        = lane & 15;
    const int khalf     = (lane >> 4) << 1;  // lanes 0-15 -> K0/K1, 16-31 -> K2/K3

    v8f acc[2][4];
    #pragma unroll
    for (int mt = 0; mt < 2; ++mt)
        #pragma unroll
        for (int nt = 0; nt < 4; ++nt)
            #pragma unroll
            for (int v = 0; v < 8; ++v) acc[mt][nt][v] = 0.0f;

    for (int kc = 0; kc < HDIM; kc += KC) {
        // ---- stage A chunk (128x32): 1024 float4, 4 per thread, coalesced ----
        #pragma unroll
        for (int i = 0; i < 4; ++i) {
            int j    = tid + i * 256;        // 0..1023
            int row  = j >> 3;               // 0..127
            int c4   = (j & 7) << 2;         // 0..28
            int grow = block_row + row;
            float4 v = make_float4(0.f, 0.f, 0.f, 0.f);
            if (grow < M)
                v = *reinterpret_cast<const float4*>(A + (size_t)grow * HDIM + kc + c4);
            if (RELU_IN) {
                v.x = fmaxf(v.x, 0.f); v.y = fmaxf(v.y, 0.f);
                v.z = fmaxf(v.z, 0.f); v.w = fmaxf(v.w, 0.f);
            }
            As[row][c4 + 0] = v.x; As[row][c4 + 1] = v.y;
            As[row][c4 + 2] = v.z; As[row][c4 + 3] = v.w;
        }
        // ---- stage B chunk (32x128) quad-packed: 1024 tasks, 4 per thread.
        //      Column-wise global reads (coalesced across lanes, L2-resident
        //      weights), one b128 LDS store per quad. ----
        #pragma unroll
        for (int i = 0; i < 4; ++i) {
            int idx  = tid + i * 256;        // 0..1023
            int col  = idx & 127;
            int quad = idx >> 7;             // 0..7 -> k = kc + quad*4 ..
            const float* bp = B + (size_t)(kc + quad * 4) * HDIM + col;
            float4 v;
            v.x = bp[0 * HDIM];
            v.y = bp[1 * HDIM];
            v.z = bp[2 * HDIM];
            v.w = bp[3 * HDIM];
            *reinterpret_cast<float4*>(&BsQ[quad][col][0]) = v;
        }
        __syncthreads();

        #pragma unroll
        for (int ks = 0; ks < KC / 4; ++ks) {
            v2f a0 = *reinterpret_cast<const v2f*>(&As[wm * 32 + lr][ks * 4 + khalf]);
            v2f a1 = *reinterpret_cast<const v2f*>(&As[wm * 32 + 16 + lr][ks * 4 + khalf]);
            #pragma unroll
            for (int nt = 0; nt < 4; ++nt) {
                int col = wn * 64 + nt * 16 + lr;
                v2f b = *reinterpret_cast<const v2f*>(&BsQ[ks][col][khalf]);
                acc[0][nt] = __builtin_amdgcn_wmma_f32_16x16x4_f32(
                    false, a0, false, b, (short)0, acc[0][nt], false, false);
                acc[1][nt] = __builtin_amdgcn_wmma_f32_16x16x4_f32(
                    false, a1, false, b, (short)0, acc[1][nt], false, false);
            }
        }
        __syncthreads();
    }

    // ---- epilogue: C/D layout = VGPR v, lanes 0-15 -> M=v, 16-31 -> M=8+v ----
    #pragma unroll
    for (int mt = 0; mt < 2; ++mt) {
        #pragma unroll
        for (int nt = 0; nt < 4; ++nt) {
            int col  = wn * 64 + nt * 16 + lr;
            float bv = BIAS ? bias[col] : 0.0f;
            #pragma unroll
            for (int v = 0; v < 8; ++v) {
                int grow = block_row + wm * 32 + mt * 16 + ((lane < 16) ? v : 8 + v);
                if (grow < M) {
                    float val = acc[mt][nt][v] + bv;
                    if (ACCUM)    val += Cin[(size_t)grow * HDIM + col];
                    if (RELU_OUT) val  = fmaxf(val, 0.0f);
                    D[(size_t)grow * HDIM + col] = val;
                }
            }
        }
    }
}

// ---------------------------------------------------------------------------
// Edge / scatter kernels
// ---------------------------------------------------------------------------
__global__ void cnt_kernel(const int* __restrict__ etype, const int* __restrict__ dst,
                           float* __restrict__ cnt, int N, int E)
{
    int e = blockIdx.x * blockDim.x + threadIdx.x;
    if (e < E) atomicAdd(&cnt[(size_t)etype[e] * N + dst[e]], 1.0f);
}

__global__ __launch_bounds__(128)
void scatter_rel_kernel(const int* __restrict__ src, const int* __restrict__ dst,
                        const int* __restrict__ etype, const float* __restrict__ cnt,
                        const float* __restrict__ msg, float* __restrict__ agg,
                        int rel, int N, int E)
{
    int e = blockIdx.x;
    if (e >= E || etype[e] != rel) return;
    int s = src[e], d = dst[e];
    float norm = 1.0f / fmaxf(cnt[(size_t)rel * N + d], 1.0f);
    int t = threadIdx.x;
    atomicAdd(&agg[(size_t)d * HDIM + t], msg[(size_t)s * HDIM + t] * norm);
}

__global__ void deg_kernel(const int* __restrict__ dst, int* __restrict__ deg, int E)
{
    int e = blockIdx.x * blockDim.x + threadIdx.x;
    if (e < E) atomicAdd(&deg[dst[e]], 1);
}

__global__ __launch_bounds__(128)
void hsum_scatter_kernel(const int* __restrict__ src, const int* __restrict__ dst,
                         const float* __restrict__ h, float* __restrict__ hsum, int E)
{
    int e = blockIdx.x;
    if (e >= E) return;
    int s = src[e], d = dst[e];
    int t = threadIdx.x;
    atomicAdd(&hsum[(size_t)d * HDIM + t], fmaxf(h[(size_t)s * HDIM + t], 0.0f));
}

// ---------------------------------------------------------------------------
// MFConv: per-node degree-selected weights. out_n = hsum_n @ Wl[deg] +
// relu(h_n) @ Wr[deg] + bl[deg]. 11 weight matrices stay L2-resident;
// weight reads are coalesced across lanes.
// ---------------------------------------------------------------------------
__global__ __launch_bounds__(128)
void mfconv_kernel(const float* __restrict__ hsum, const float* __restrict__ h,
                   const int* __restrict__ deg,
                   const float* __restrict__ Wl, const float* __restrict__ Wr,
                   const float* __restrict__ bl, float* __restrict__ out, int N)
{
    int n = blockIdx.x;
    if (n >= N) return;
    int t = threadIdx.x;
    int d = min(deg[n], 10);

    __shared__ float sh[HDIM];
    __shared__ float sx[HDIM];
    sh[t] = hsum[(size_t)n * HDIM + t];
    sx[t] = fmaxf(h[(size_t)n * HDIM + t], 0.0f);
    __syncthreads();

    const float* Wld = Wl + (size_t)d * HDIM * HDIM;
    const float* Wrd = Wr + (size_t)d * HDIM * HDIM;
    float acc = bl[(size_t)d * HDIM + t];
    #pragma unroll 4
    for (int f = 0; f < HDIM; ++f)
        acc += sh[f] * Wld[(size_t)f * HDIM + t] + sx[f] * Wrd[(size_t)f * HDIM + t];
    out[(size_t)n * HDIM + t] = acc;
}

// ---------------------------------------------------------------------------
// Pool + head
// ---------------------------------------------------------------------------
__global__ __launch_bounds__(128)
void pool_kernel(const float* __restrict__ h, const int* __restrict__ batch,
                 float* __restrict__ pooled, int N)
{
    int n = blockIdx.x;
    if (n >= N) return;
    int t = threadIdx.x;
    atomicAdd(&pooled[(size_t)batch[n] * HDIM + t], h[(size_t)n * HDIM + t]);
}

__global__ __launch_bounds__(128)
void head1_kernel(const float* __restrict__ pooled, const float* __restrict__ W1,
                  const float* __restrict__ b1, float* __restrict__ z)
{
    int g = blockIdx.x;
    int t = threadIdx.x;
    __shared__ float sp[HDIM];
    sp[t] = pooled[(size_t)g * HDIM + t];
    __syncthreads();
    float acc = b1[t];
    #pragma unroll 4
    for (int f = 0; f < HDIM; ++f) acc += sp[f] * W1[(size_t)f * HDIM + t];
    z[(size_t)g * HDIM + t] = fmaxf(acc, 0.0f);
}

__global__ void head2_kernel(const float* __restrict__ z, const float* __restrict__ W2,
                             const float* __restrict__ b2, float* __restrict__ out,
                             int total)  // total = G*OUT
{
    int i = blockIdx.x * blockDim.x + threadIdx.x;
    if (i >= total) return;
    int g = i / 10, o = i % 10;
    float acc = b2[o];
    #pragma unroll 4
    for (int f = 0; f < HDIM; ++f) acc += z[(size_t)g * HDIM + f] * W2[(size_t)f * 10 + o];
    out[(size_t)g * 10 + o] = acc;
}

// ---------------------------------------------------------------------------
// Orchestration
// ---------------------------------------------------------------------------
extern "C" void kernel_launch(void* const* d_in, const int* in_sizes, int n_in,
                              void* d_out, int out_size, void* d_ws, size_t ws_size,
                              hipStream_t stream)
{
    const float* x        = (const float*)d_in[0];
    const int*   ei       = (const int*)  d_in[1];
    const int*   etype    = (const int*)  d_in[2];
    const int*   batch    = (const int*)  d_in[3];
    const float* emb_W    = (const float*)d_in[4];
    const float* emb_b    = (const float*)d_in[5];
    const float* rgcn_W   = (const float*)d_in[6];   // [NB,R,H,H]
    const float* rgcn_root= (const float*)d_in[7];   // [NB,H,H]
    const float* rgcn_b   = (const float*)d_in[8];   // [NB,H]
    const float* mf_Wl    = (const float*)d_in[9];   // [NB,KD,H,H]
    const float* mf_bl    = (const float*)d_in[10];  // [NB,KD,H]
    const float* mf_Wr    = (const float*)d_in[11];  // [NB,KD,H,H]
    const float* head_W1  = (const float*)d_in[12];
    const float* head_b1  = (const float*)d_in[13];
    const float* head_W2  = (const float*)d_in[14];
    const float* head_b2  = (const float*)d_in[15];
    float* out = (float*)d_out;

    const int N = in_sizes[0] / HDIM;     // 30000
    const int E = in_sizes[1] / 2;        // 480000
    const int R = 8, KD = 11, G = 64;
    const int* src = ei;
    const int* dst = ei + E;

    // workspace layout (all 16B-aligned: N*H*4 is a multiple of 16)
    const size_t NH = (size_t)N * HDIM;
    float* h0     = (float*)d_ws;
    float* h1     = h0  + NH;
    float* tmp    = h1  + NH;            // per-relation msg / MF neighbor sum
    float* agg    = tmp + NH;
    float* cnt    = agg + NH;            // R*N floats
    int*   deg    = (int*)(cnt + (size_t)R * N);
    float* pooled = (float*)(deg + N);
    float* zbuf   = pooled + (size_t)G * HDIM;

    const int gemmGrid = (N + BM - 1) / BM;
    const int eGrid256 = (E + 255) / 256;

    // 1) embed: h0 = x @ emb_W + emb_b
    gemm128_wmma<false, true, false, false>
        <<<gemmGrid, 256, 0, stream>>>(x, emb_W, emb_b, nullptr, h0, N);

    for (int i = 0; i < 2; ++i) {
        // ---- RGCN layer (input relu fused into GEMM A-reads) ----
        hipMemsetAsync(cnt, 0, (size_t)R * N * sizeof(float), stream);
        hipMemsetAsync(agg, 0, NH * sizeof(float), stream);
        cnt_kernel<<<eGrid256, 256, 0, stream>>>(etype, dst, cnt, N, E);

        for (int r = 0; r < R; ++r) {
            const float* Wr_ = rgcn_W + (size_t)(i * R + r) * HDIM * HDIM;
            gemm128_wmma<true, false, false, false>
                <<<gemmGrid, 256, 0, stream>>>(h0, Wr_, nullptr, nullptr, tmp, N);
            scatter_rel_kernel<<<E, 128, 0, stream>>>(src, dst, etype, cnt, tmp, agg, r, N, E);
        }
        // root transform + bias + aggregated messages -> h1
        gemm128_wmma<true, true, true, false>
            <<<gemmGrid, 256, 0, stream>>>(h0, rgcn_root + (size_t)i * HDIM * HDIM,
                                           rgcn_b + (size_t)i * HDIM, agg, h1, N);

        // ---- MFConv layer (input relu fused into scatter + node kernel) ----
        hipMemsetAsync(deg, 0, (size_t)N * sizeof(int), stream);
        hipMemsetAsync(tmp, 0, NH * sizeof(float), stream);
        deg_kernel<<<eGrid256, 256, 0, stream>>>(dst, deg, E);
        hsum_scatter_kernel<<<E, 128, 0, stream>>>(src, dst, h1, tmp, E);
        mfconv_kernel<<<N, 128, 0, stream>>>(tmp, h1, deg,
            mf_Wl + (size_t)i * KD * HDIM * HDIM,
            mf_Wr + (size_t)i * KD * HDIM * HDIM,
            mf_bl + (size_t)i * KD * HDIM, h0, N);
    }

    // ---- pool + head ----
    hipMemsetAsync(pooled, 0, (size_t)G * HDIM * sizeof(float), stream);
    pool_kernel<<<N, 128, 0, stream>>>(h0, batch, pooled, N);
    head1_kernel<<<G, 128, 0, stream>>>(pooled, head_W1, head_b1, zbuf);
    head2_kernel<<<(G * 10 + 127) / 128, 128, 0, stream>>>(zbuf, head_W2, head_b2, out, G * 10);
}